// EmbeddingD_17755394802312
// MI455X (gfx1250) — compile-verified
//
#include <hip/hip_runtime.h>

#define FDL 1024
#define NN  884
#define EE  100000
#define NE  (NN * FDL)

typedef __attribute__((ext_vector_type(2))) float v2f;
typedef __attribute__((ext_vector_type(8))) float v8f;

__device__ __forceinline__ v8f wmma4(v2f a, v2f b, v8f c) {
  // D = A(16x4,f32) x B(4x16,f32) + C(16x16,f32)
  return __builtin_amdgcn_wmma_f32_16x16x4_f32(false, a, false, b, (short)0, c,
                                               false, false);
}

// ---------------------------------------------------------------- utilities
__global__ void fill_f32(float* __restrict__ p, float v, int n) {
  int i = blockIdx.x * blockDim.x + threadIdx.x;
  if (i < n) p[i] = v;
}

__global__ void zero_i32(int* __restrict__ p, int n) {
  int i = blockIdx.x * blockDim.x + threadIdx.x;
  if (i < n) p[i] = 0;
}

// deg[dst] += ew   (deg pre-initialized to 1.0 for the self loop)
__global__ void deg_scatter(const int* __restrict__ dst,
                            const float* __restrict__ ew,
                            float* __restrict__ deg, int e) {
  int i = blockIdx.x * blockDim.x + threadIdx.x;
  if (i < e) atomicAdd(&deg[dst[i]], ew[i]);
}

__global__ void rsqrt_inplace(float* __restrict__ d, int n) {
  int i = blockIdx.x * blockDim.x + threadIdx.x;
  if (i < n) d[i] = rsqrtf(d[i]);
}

__global__ void hist_kernel(const int* __restrict__ dst, int* __restrict__ hist,
                            int e) {
  int i = blockIdx.x * blockDim.x + threadIdx.x;
  if (i < e) atomicAdd(&hist[dst[i]], 1);
}

// exclusive scan of hist[0..n-1] -> offs[0..n]; cursor = offs copy.
// hist and cursor may alias (single block, reads complete before writes).
__global__ __launch_bounds__(1024) void scan_kernel(const int* __restrict__ hist,
                                                    int* __restrict__ offs,
                                                    int* __restrict__ cursor,
                                                    int n) {
  __shared__ int buf[2][1024];
  int tid = threadIdx.x;
  int v = (tid < n) ? hist[tid] : 0;
  buf[0][tid] = v;
  __syncthreads();
  int src = 0;
  for (int off = 1; off < 1024; off <<= 1) {
    int x = buf[src][tid];
    if (tid >= off) x += buf[src][tid - off];
    buf[src ^ 1][tid] = x;
    __syncthreads();
    src ^= 1;
  }
  int incl = buf[src][tid];
  int excl = incl - v;
  if (tid < n) {
    offs[tid] = excl;
    cursor[tid] = excl;
  }
  if (tid == n - 1) offs[n] = incl;
}

// counting-sort placement: sorted src + normalized weight per edge
__global__ void place_kernel(const int* __restrict__ src,
                             const int* __restrict__ dst,
                             const float* __restrict__ ew,
                             const float* __restrict__ dinv,
                             int* __restrict__ cursor, int* __restrict__ ssrc,
                             float* __restrict__ sw, int e) {
  int i = blockIdx.x * blockDim.x + threadIdx.x;
  if (i >= e) return;
  int s = src[i], d = dst[i];
  float w = dinv[s] * ew[i] * dinv[d];
  int pos = atomicAdd(&cursor[d], 1);
  ssrc[pos] = s;
  sw[pos] = w;
}

// ---------------------------------------------------------------- WMMA GEMM
// C[M x 1024] = A[M x 1024] @ B[1024 x 1024], all f32 row-major.
// Block: 256 threads = 8 waves. Wave = 32 rows x 64 cols (2 m-tiles x 4
// n-tiles, 8 accumulators => B fragment reused across 2 WMMAs).
// Block = 128 rows x 128 cols. Grid = (1024/128, ceil(M/128)).
__global__ __launch_bounds__(256) void gemm_f32_wmma(const float* __restrict__ A,
                                                     const float* __restrict__ B,
                                                     float* __restrict__ C,
                                                     int M) {
  const int lane = threadIdx.x & 31;
  const int wv   = threadIdx.x >> 5;
  const int half = lane >> 4;   // 0: lanes 0-15, 1: lanes 16-31
  const int r16  = lane & 15;
  const int mg   = wv & 3;      // 4 m-groups (32 rows each) per block
  const int ng   = wv >> 2;     // 2 n-groups (64 cols each) per block

  const int m_base = blockIdx.y * 128 + mg * 32;
  const int n_base = blockIdx.x * 128 + ng * 64;

  int arow0 = m_base + r16;
  int arow1 = m_base + 16 + r16;
  if (arow0 >= M) arow0 = M - 1;  // clamp loads; stores guarded below
  if (arow1 >= M) arow1 = M - 1;

  // A fragment: lane holds A[row][k + 2*half .. +1] -> contiguous float2
  const float* Ap0 = A + (size_t)arow0 * FDL + 2 * half;
  const float* Ap1 = A + (size_t)arow1 * FDL + 2 * half;
  // B fragment: lane holds B[k + 2*half][col], B[k + 2*half + 1][col]
  const float* Bp = B + (size_t)(2 * half) * FDL + n_base + r16;

  v8f acc0 = {}, acc1 = {}, acc2 = {}, acc3 = {};
  v8f acc4 = {}, acc5 = {}, acc6 = {}, acc7 = {};

#pragma unroll 2
  for (int k = 0; k < FDL; k += 4) {
    v2f a0 = *(const v2f*)(Ap0 + k);
    v2f a1 = *(const v2f*)(Ap1 + k);
    const float* b0 = Bp + (size_t)k * FDL;
    v2f bf0, bf1, bf2, bf3;
    bf0.x = b0[0];   bf0.y = b0[FDL];
    bf1.x = b0[16];  bf1.y = b0[FDL + 16];
    bf2.x = b0[32];  bf2.y = b0[FDL + 32];
    bf3.x = b0[48];  bf3.y = b0[FDL + 48];
    acc0 = wmma4(a0, bf0, acc0);
    acc4 = wmma4(a1, bf0, acc4);
    acc1 = wmma4(a0, bf1, acc1);
    acc5 = wmma4(a1, bf1, acc5);
    acc2 = wmma4(a0, bf2, acc2);
    acc6 = wmma4(a1, bf2, acc6);
    acc3 = wmma4(a0, bf3, acc3);
    acc7 = wmma4(a1, bf3, acc7);
  }

  // C/D layout: VGPR i -> row tile_base + i + 8*half, col n_base + r16
#pragma unroll
  for (int i = 0; i < 8; ++i) {
    int row0 = m_base + i + 8 * half;
    if (row0 < M) {
      float* Cr = C + (size_t)row0 * FDL + n_base + r16;
      Cr[0]  = acc0[i];
      Cr[16] = acc1[i];
      Cr[32] = acc2[i];
      Cr[48] = acc3[i];
    }
    int row1 = m_base + 16 + i + 8 * half;
    if (row1 < M) {
      float* Cr = C + (size_t)row1 * FDL + n_base + r16;
      Cr[0]  = acc4[i];
      Cr[16] = acc5[i];
      Cr[32] = acc6[i];
      Cr[48] = acc7[i];
    }
  }
}

// -------------------------------------------------------- CSR row gather
// out[r][:] = relu( dinv[r]^2 * xw[r][:] + sum_j sw[j]*xw[ssrc[j]][:] + b )
// One block per row; 256 threads x float4 = full 1024-feature row.
#define AGG_CHUNK 256
__global__ __launch_bounds__(256) void agg_kernel(const float4* __restrict__ xw4,
                                                  const int* __restrict__ ssrc,
                                                  const float* __restrict__ sw,
                                                  const int* __restrict__ offs,
                                                  const float* __restrict__ dinv,
                                                  const float4* __restrict__ bias4,
                                                  float4* __restrict__ out4) {
  __shared__ int   s_src[AGG_CHUNK];
  __shared__ float s_w[AGG_CHUNK];
  const int r = blockIdx.x;
  const int t = threadIdx.x;
  const float di = dinv[r];
  const float selfw = di * di;
  float4 v = xw4[(size_t)r * 256 + t];
  float4 acc;
  acc.x = selfw * v.x;
  acc.y = selfw * v.y;
  acc.z = selfw * v.z;
  acc.w = selfw * v.w;

  const int e0 = offs[r], e1 = offs[r + 1];
  for (int base = e0; base < e1; base += AGG_CHUNK) {
    const int cnt = min(AGG_CHUNK, e1 - base);
    if (t < cnt) {
      s_src[t] = ssrc[base + t];
      s_w[t]   = sw[base + t];
    }
    if (t == 0 && base + AGG_CHUNK < e1) {
      __builtin_prefetch(&ssrc[base + AGG_CHUNK], 0, 0);
      __builtin_prefetch(&sw[base + AGG_CHUNK], 0, 0);
    }
    __syncthreads();
    for (int j = 0; j < cnt; ++j) {
      const float w = s_w[j];
      const float4 xv = xw4[(size_t)s_src[j] * 256 + t];
      acc.x = fmaf(w, xv.x, acc.x);
      acc.y = fmaf(w, xv.y, acc.y);
      acc.z = fmaf(w, xv.z, acc.z);
      acc.w = fmaf(w, xv.w, acc.w);
    }
    __syncthreads();
  }

  const float4 b = bias4[t];
  float4 o;
  o.x = fmaxf(acc.x + b.x, 0.0f);
  o.y = fmaxf(acc.y + b.y, 0.0f);
  o.z = fmaxf(acc.z + b.z, 0.0f);
  o.w = fmaxf(acc.w + b.w, 0.0f);
  out4[(size_t)r * 256 + t] = o;
}

// ------------------------------------------------------------ SE epilogue
__global__ __launch_bounds__(256) void mean_kernel(const float4* __restrict__ h4,
                                                   float* __restrict__ means,
                                                   int ne4) {
  const int c = blockIdx.y;
  const float4* p = h4 + (size_t)c * ne4;
  int base = blockIdx.x * (256 * 16) + threadIdx.x;
  float s = 0.0f;
#pragma unroll
  for (int i = 0; i < 16; ++i) {
    int idx = base + i * 256;
    if (idx < ne4) {
      float4 v = p[idx];
      s += (v.x + v.y) + (v.z + v.w);
    }
  }
  __shared__ float red[256];
  red[threadIdx.x] = s;
  __syncthreads();
  for (int o = 128; o > 0; o >>= 1) {
    if (threadIdx.x < o) red[threadIdx.x] += red[threadIdx.x + o];
    __syncthreads();
  }
  if (threadIdx.x == 0) atomicAdd(&means[c], red[0]);
}

__global__ void se_kernel(const float* __restrict__ means,
                          const float* __restrict__ fc1w,
                          const float* __restrict__ fc1b,
                          const float* __restrict__ fc2w,
                          const float* __restrict__ fc2b,
                          float* __restrict__ sca, float inv_ne) {
  if (threadIdx.x != 0 || blockIdx.x != 0) return;
  float m[6], a[30];
  for (int c = 0; c < 6; ++c) m[c] = means[c] * inv_ne;
  for (int j = 0; j < 30; ++j) {
    float t = fc1b[j];
    for (int c = 0; c < 6; ++c) t += m[c] * fc1w[c * 30 + j];
    a[j] = fmaxf(t, 0.0f);
  }
  for (int c = 0; c < 6; ++c) {
    float t = fc2b[c];
    for (int j = 0; j < 30; ++j) t += a[j] * fc2w[j * 6 + c];
    sca[c] = 1.0f / (1.0f + expf(-t));
  }
}

__global__ void final_kernel(const float4* __restrict__ h4,
                             const float* __restrict__ sca,
                             const float* __restrict__ cw,
                             const float* __restrict__ cb,
                             float4* __restrict__ out4, int ne4) {
  int idx = blockIdx.x * blockDim.x + threadIdx.x;
  if (idx >= ne4) return;
  const float b = cb[0];
  float4 acc = {b, b, b, b};
#pragma unroll
  for (int c = 0; c < 6; ++c) {
    const float s = sca[c];
    const float w = cw[c];
    float4 v = h4[(size_t)c * ne4 + idx];
    acc.x += w * fmaxf(s * v.x, 0.0f);
    acc.y += w * fmaxf(s * v.y, 0.0f);
    acc.z += w * fmaxf(s * v.z, 0.0f);
    acc.w += w * fmaxf(s * v.w, 0.0f);
  }
  out4[idx] = acc;
}

// ------------------------------------------------------------------ driver
extern "C" void kernel_launch(void* const* d_in, const int* in_sizes, int n_in,
                              void* d_out, int out_size, void* d_ws,
                              size_t ws_size, hipStream_t stream) {
  (void)in_sizes; (void)n_in; (void)out_size; (void)ws_size;

  const float* x_d = (const float*)d_in[0];
  const int*   srcv[3] = {(const int*)d_in[1], (const int*)d_in[4], (const int*)d_in[7]};
  const int*   dstv[3] = {(const int*)d_in[2], (const int*)d_in[5], (const int*)d_in[8]};
  const float* ewv[3]  = {(const float*)d_in[3], (const float*)d_in[6], (const float*)d_in[9]};
  const float* Wv[6];
  const float* bv[6];
  for (int i = 0; i < 6; ++i) {
    Wv[i] = (const float*)d_in[10 + 2 * i];
    bv[i] = (const float*)d_in[11 + 2 * i];
  }
  const float* fc1w = (const float*)d_in[22];
  const float* fc1b = (const float*)d_in[23];
  const float* fc2w = (const float*)d_in[24];
  const float* fc2b = (const float*)d_in[25];
  const float* cnnw = (const float*)d_in[26];
  const float* cnnb = (const float*)d_in[27];

  // ---- workspace carve (256B aligned)
  char* ws = (char*)d_ws;
  auto carve = [&](size_t bytes) {
    void* p = (void*)ws;
    ws += (bytes + 255) & ~(size_t)255;
    return p;
  };
  float* hbuf   = (float*)carve((size_t)6 * NE * 4);  // t1,t2,s1,s2,g1,g2
  float* xw     = (float*)carve((size_t)NE * 4);      // x @ W scratch
  float* dinv   = (float*)carve((size_t)NN * 4);      // deg -> dinv in place
  int*   ssrc   = (int*)carve((size_t)EE * 4);        // sorted src
  float* sw     = (float*)carve((size_t)EE * 4);      // sorted norm weights
  int*   offs   = (int*)carve((size_t)(NN + 1) * 4);  // CSR offsets
  int*   cursor = (int*)carve((size_t)NN * 4);        // hist / placement cursor
  float* means  = (float*)carve(6 * 4);
  float* sca    = (float*)carve(6 * 4);

  const int  ne4 = NE / 4;                            // 226304
  const dim3 gemm_grid(FDL / 128, (NN + 127) / 128);  // (8, 7)
  const int  eb = (EE + 255) / 256;
  const int  nb = (NN + 255) / 256;

  for (int v = 0; v < 3; ++v) {
    // degree -> dinv (self loop weight 1 => deg starts at 1)
    fill_f32<<<nb, 256, 0, stream>>>(dinv, 1.0f, NN);
    deg_scatter<<<eb, 256, 0, stream>>>(dstv[v], ewv[v], dinv, EE);
    rsqrt_inplace<<<nb, 256, 0, stream>>>(dinv, NN);
    // counting sort of edges by destination
    zero_i32<<<nb, 256, 0, stream>>>(cursor, NN);
    hist_kernel<<<eb, 256, 0, stream>>>(dstv[v], cursor, EE);
    scan_kernel<<<1, 1024, 0, stream>>>(cursor, offs, cursor, NN);
    place_kernel<<<eb, 256, 0, stream>>>(srcv[v], dstv[v], ewv[v], dinv, cursor,
                                         ssrc, sw, EE);
    // two GCN layers
    const float* hin = x_d;
    for (int l = 0; l < 2; ++l) {
      const int ci = v * 2 + l;
      float* hout = hbuf + (size_t)ci * NE;
      gemm_f32_wmma<<<gemm_grid, 256, 0, stream>>>(hin, Wv[ci], xw, NN);
      agg_kernel<<<NN, 256, 0, stream>>>((const float4*)xw, ssrc, sw, offs,
                                         dinv, (const float4*)bv[ci],
                                         (float4*)hout);
      hin = hout;
    }
  }

  // SE channel attention + 1x1 conv
  fill_f32<<<1, 32, 0, stream>>>(means, 0.0f, 6);
  mean_kernel<<<dim3((ne4 + 4095) / 4096, 6), 256, 0, stream>>>(
      (const float4*)hbuf, means, ne4);
  se_kernel<<<1, 32, 0, stream>>>(means, fc1w, fc1b, fc2w, fc2b, sca,
                                  1.0f / (float)NE);
  final_kernel<<<(ne4 + 255) / 256, 256, 0, stream>>>(
      (const float4*)hbuf, sca, cnnw, cnnb, (float4*)d_out, ne4);
}